// BettingLoss_87076166960015
// MI455X (gfx1250) — compile-verified
//
#include <hip/hip_runtime.h>
#include <stddef.h>

// ---------------- constants ----------------
constexpr int   TT    = 6;                      // traps per race
constexpr int   NBLK  = 2048;                   // blocks in kernel 1
constexpr int   TPB   = 256;                    // threads per block
constexpr int   TILE  = 256;                    // races per tile
constexpr int   TILES = 8;                      // tiles per block
constexpr int   RACES_PER_BLOCK = TILE * TILES; // 2048
constexpr int   ARR_BYTES = TILE * TT * 4;      // 6144 bytes per array per tile
constexpr float BF = 4194304.0f;                // batch size as float

// async path only if the gfx1250 async-to-LDS builtin exists on this toolchain
#if defined(__gfx1250__) && __has_builtin(__builtin_amdgcn_global_load_async_to_lds_b64)
#define HAVE_ASYNC 1
#else
#define HAVE_ASYNC 0
#endif

#if HAVE_ASYNC
// builtin proto: void(v2i as1* src, v2i as3* dst, Ii off, Ii cpol)
typedef int v2i_async __attribute__((vector_size(8)));
typedef __attribute__((address_space(1))) v2i_async* g_v2i_ptr;
typedef __attribute__((address_space(3))) v2i_async* l_v2i_ptr;
#if __has_builtin(__builtin_amdgcn_s_wait_asynccnt)
#define WAIT_ASYNCCNT(n) __builtin_amdgcn_s_wait_asynccnt(n)
#else
#define WAIT_ASYNCCNT(n) asm volatile("s_wait_asynccnt %0" ::"n"(n) : "memory")
#endif
#endif

// WMMA-based final reduction (f32 16x16x4): with B==1 and C==0 the result's
// row sums cover every A element exactly once -> layout-independent total.
#if defined(__gfx1250__) && __has_builtin(__builtin_amdgcn_wmma_f32_16x16x4_f32)
#define HAVE_WMMA 1
#else
#define HAVE_WMMA 0
#endif

typedef float v2f __attribute__((ext_vector_type(2)));
typedef float v8f __attribute__((ext_vector_type(8)));

// ---------------- per-race compute (matches reference exactly) ----------------
__device__ __forceinline__ void race_accum(const float p[6], const float w[6],
                                           const float o[6], float& accEp,
                                           float& accProfit, float& accMax,
                                           unsigned& accNb) {
  bool hasValid = false;
#pragma unroll
  for (int t = 0; t < 6; ++t) hasValid = hasValid || (o[t] > 0.0f);

  float ep[6];
#pragma unroll
  for (int t = 0; t < 6; ++t) {
    float v = (o[t] * 1.1f * p[t] - 1.0f) * 0.019f;
    ep[t] = hasValid ? v : 0.0f;
  }

  // first-max argmax (strict > keeps first occurrence, like jnp.argmax)
  float be = ep[0], ob = o[0], wb = w[0];
#pragma unroll
  for (int t = 1; t < 6; ++t) {
    bool g = ep[t] > be;
    be = g ? ep[t] : be;
    ob = g ? o[t] : ob;
    wb = g ? w[t] : wb;
  }

  bool bet = (be > 0.0f) && hasValid;
  float winp  = (ob * 1.1f * 0.02f - 0.02f) * 0.95f;
  float losep = -0.02f * 0.95f;

  accEp     += bet ? be : 0.0f;
  accProfit += bet ? (wb > 0.5f ? winp : losep) : 0.0f;
  accNb     += bet ? 1u : 0u;

  float mx = p[0];
#pragma unroll
  for (int t = 1; t < 6; ++t) mx = fmaxf(mx, p[t]);
  accMax += mx;
}

// ---------------- kernel 1: streaming partial reduction ----------------
__global__ __launch_bounds__(TPB) void bet_partial(
    const float* __restrict__ probs, const float* __restrict__ winners,
    const float* __restrict__ odds, float* __restrict__ wsf,
    unsigned* __restrict__ wsn) {
  const int tid = threadIdx.x;
  const int blk = blockIdx.x;

  float accEp = 0.0f, accProfit = 0.0f, accMax = 0.0f;
  unsigned accNb = 0u;

#if HAVE_ASYNC
  // double-buffered async global->LDS DMA pipeline (ASYNCcnt-tracked)
  __shared__ __align__(16) unsigned char lds_raw[2][3 * ARR_BYTES];
  const char* srcs[3] = {(const char*)probs, (const char*)winners,
                         (const char*)odds};

  auto issue = [&](int tile) {
    size_t byteOff =
        ((size_t)blk * RACES_PER_BLOCK + (size_t)tile * TILE) * (TT * 4);
    unsigned char* buf = lds_raw[tile & 1];
#pragma unroll
    for (int a = 0; a < 3; ++a) {
      const char* s = srcs[a] + byteOff;
      unsigned char* d = buf + a * ARR_BYTES;
#pragma unroll
      for (int q = 0; q < 3; ++q) {
        int off = q * 2048 + tid * 8;  // 768 x b64 per array, 3 per thread
        __builtin_amdgcn_global_load_async_to_lds_b64(
            (g_v2i_ptr)(void*)(s + off), (l_v2i_ptr)(void*)(d + off), 0, 0);
      }
    }
  };

  issue(0);
  for (int tile = 0; tile < TILES; ++tile) {
    if (tile + 1 < TILES) {
      issue(tile + 1);
      WAIT_ASYNCCNT(9);  // 9 newest outstanding -> current tile's 9 complete
    } else {
      WAIT_ASYNCCNT(0);
    }
    __syncthreads();  // all waves' DMA for this tile visible

    const unsigned char* buf = lds_raw[tile & 1];
    const float* bp = (const float*)(buf);
    const float* bw = (const float*)(buf + ARR_BYTES);
    const float* bo = (const float*)(buf + 2 * ARR_BYTES);
    const int row = tid * TT;
    float p[6], w[6], o[6];
#pragma unroll
    for (int t = 0; t < 6; ++t) {
      p[t] = bp[row + t];
      w[t] = bw[row + t];
      o[t] = bo[row + t];
    }
    race_accum(p, w, o, accEp, accProfit, accMax, accNb);
    __syncthreads();  // safe to overwrite this buffer two tiles later
  }
#else
  // fallback: direct non-temporal vectorized loads
  for (int tile = 0; tile < TILES; ++tile) {
    size_t race = (size_t)blk * RACES_PER_BLOCK + (size_t)tile * TILE + tid;
    const float* pb = probs + race * TT;
    const float* wbp = winners + race * TT;
    const float* ob = odds + race * TT;
    v2f p01 = __builtin_nontemporal_load((const v2f*)(pb + 0));
    v2f p23 = __builtin_nontemporal_load((const v2f*)(pb + 2));
    v2f p45 = __builtin_nontemporal_load((const v2f*)(pb + 4));
    v2f w01 = __builtin_nontemporal_load((const v2f*)(wbp + 0));
    v2f w23 = __builtin_nontemporal_load((const v2f*)(wbp + 2));
    v2f w45 = __builtin_nontemporal_load((const v2f*)(wbp + 4));
    v2f o01 = __builtin_nontemporal_load((const v2f*)(ob + 0));
    v2f o23 = __builtin_nontemporal_load((const v2f*)(ob + 2));
    v2f o45 = __builtin_nontemporal_load((const v2f*)(ob + 4));
    float p[6] = {p01.x, p01.y, p23.x, p23.y, p45.x, p45.y};
    float w[6] = {w01.x, w01.y, w23.x, w23.y, w45.x, w45.y};
    float o[6] = {o01.x, o01.y, o23.x, o23.y, o45.x, o45.y};
    race_accum(p, w, o, accEp, accProfit, accMax, accNb);
  }
#endif

  // fixed-order block tree reduction (deterministic)
  __shared__ float s_e[TPB], s_p[TPB], s_m[TPB];
  __shared__ unsigned s_n[TPB];
  s_e[tid] = accEp;
  s_p[tid] = accProfit;
  s_m[tid] = accMax;
  s_n[tid] = accNb;
  __syncthreads();
  for (int off = TPB / 2; off > 0; off >>= 1) {
    if (tid < off) {
      s_e[tid] += s_e[tid + off];
      s_p[tid] += s_p[tid + off];
      s_m[tid] += s_m[tid + off];
      s_n[tid] += s_n[tid + off];
    }
    __syncthreads();
  }
  if (tid == 0) {
    wsf[blk] = s_e[0];
    wsf[NBLK + blk] = s_p[0];
    wsf[2 * NBLK + blk] = s_m[0];
    wsn[blk] = s_n[0];
  }
}

#if HAVE_WMMA
// Sum 64 floats held by wave 0 (a.x/a.y per lane) with one WMMA:
// D = A x Ones + 0 -> D[m][n] = rowsum(A,m). Lane 0 VGPRs hold rows 0..7,
// lane 16 VGPRs hold rows 8..15 (documented C/D layout). Total = sum of all A.
__device__ __forceinline__ float wmma_sum64(v2f a) {
  v2f ones = {1.0f, 1.0f};
  v8f zero = {};
  v8f d = __builtin_amdgcn_wmma_f32_16x16x4_f32(false, a, false, ones,
                                                (short)0, zero, false, false);
  float r = 0.0f;
#pragma unroll
  for (int j = 0; j < 8; ++j) r += d[j];
  return r + __shfl(r, 16);  // lane 0: rows 0..7 + rows 8..15
}
#endif

// ---------------- kernel 2: finalize ----------------
__global__ __launch_bounds__(TPB) void bet_final(const float* __restrict__ wsf,
                                                 const unsigned* __restrict__ wsn,
                                                 float* __restrict__ out) {
  const int tid = threadIdx.x;
  float accEp = 0.0f, accProfit = 0.0f, accMax = 0.0f, accNb = 0.0f;
  for (int i = tid; i < NBLK; i += TPB) {  // fixed-order strided accumulation
    accEp += wsf[i];
    accProfit += wsf[NBLK + i];
    accMax += wsf[2 * NBLK + i];
    accNb += (float)wsn[i];  // exact: all counts < 2^24
  }
  __shared__ float s_e[TPB], s_p[TPB], s_m[TPB], s_n[TPB];
  s_e[tid] = accEp;
  s_p[tid] = accProfit;
  s_m[tid] = accMax;
  s_n[tid] = accNb;
  __syncthreads();
  // tree reduce 256 -> 64
  if (tid < 128) {
    s_e[tid] += s_e[tid + 128];
    s_p[tid] += s_p[tid + 128];
    s_m[tid] += s_m[tid + 128];
    s_n[tid] += s_n[tid + 128];
  }
  __syncthreads();
  if (tid < 64) {
    s_e[tid] += s_e[tid + 64];
    s_p[tid] += s_p[tid + 64];
    s_m[tid] += s_m[tid + 64];
    s_n[tid] += s_n[tid + 64];
  }
  __syncthreads();
#if HAVE_WMMA
  if (tid < 32) {  // wave 0, EXEC all ones across its 32 lanes
    v2f ae = {s_e[tid], s_e[tid + 32]};
    v2f ap = {s_p[tid], s_p[tid + 32]};
    v2f am = {s_m[tid], s_m[tid + 32]};
    v2f an = {s_n[tid], s_n[tid + 32]};
    float totE = wmma_sum64(ae);
    float totP = wmma_sum64(ap);
    float totM = wmma_sum64(am);
    float totN = wmma_sum64(an);
    if (tid == 0) {
      float total = (totN > 0.0f) ? totE : (-(totM / BF) * 0.1f);
      out[0] = -total / BF;  // loss
      out[1] = totP;         // batch_profit
      out[2] = totN;         // num_bets (exact integer in fp32)
    }
  }
#else
  for (int off = 32; off > 0; off >>= 1) {
    if (tid < off) {
      s_e[tid] += s_e[tid + off];
      s_p[tid] += s_p[tid + off];
      s_m[tid] += s_m[tid + off];
      s_n[tid] += s_n[tid + off];
    }
    __syncthreads();
  }
  if (tid == 0) {
    float total = (s_n[0] > 0.0f) ? s_e[0] : (-(s_m[0] / BF) * 0.1f);
    out[0] = -total / BF;
    out[1] = s_p[0];
    out[2] = s_n[0];
  }
#endif
}

// ---------------- launch ----------------
extern "C" void kernel_launch(void* const* d_in, const int* in_sizes, int n_in,
                              void* d_out, int out_size, void* d_ws,
                              size_t ws_size, hipStream_t stream) {
  const float* probs = (const float*)d_in[0];
  const float* winners = (const float*)d_in[1];
  const float* odds = (const float*)d_in[2];
  float* wsf = (float*)d_ws;                    // 3*NBLK floats
  unsigned* wsn = (unsigned*)(wsf + 3 * NBLK);  // NBLK uints
  bet_partial<<<NBLK, TPB, 0, stream>>>(probs, winners, odds, wsf, wsn);
  bet_final<<<1, TPB, 0, stream>>>(wsf, wsn, (float*)d_out);
}